// GNN_15616501088474
// MI455X (gfx1250) — compile-verified
//
#include <hip/hip_runtime.h>

#define D 128
#define BN_EPS 1e-5f

typedef __attribute__((ext_vector_type(2))) float v2f;
typedef __attribute__((ext_vector_type(8))) float v8f;

static __device__ __forceinline__ void atomic_add_f32(float* p, float v) {
    __hip_atomic_fetch_add(p, v, __ATOMIC_RELAXED, __HIP_MEMORY_SCOPE_AGENT);
}

// ================= CSR construction =================
__global__ void k_zero_i32(int* __restrict__ a, int len) {
    int i = blockIdx.x * blockDim.x + threadIdx.x;
    if (i < len) a[i] = 0;
}

// in-degree counts into rowptr[d+1] so the inclusive scan yields row offsets
__global__ void k_count(const long long* __restrict__ dst, int* __restrict__ rowptr, int e) {
    int i = blockIdx.x * blockDim.x + threadIdx.x;
    if (i < e) atomicAdd(&rowptr[(int)dst[i] + 1], 1);
}

// single-workgroup chunked inclusive scan (len ~ 1e5 -> ~98 chunk iterations)
__global__ __launch_bounds__(1024) void k_scan(int* __restrict__ a, int len) {
    __shared__ int buf[1024];
    __shared__ int carry;
    if (threadIdx.x == 0) carry = 0;
    __syncthreads();
    for (int base = 0; base < len; base += 1024) {
        int i = base + threadIdx.x;
        int v = (i < len) ? a[i] : 0;
        buf[threadIdx.x] = v;
        __syncthreads();
        for (int off = 1; off < 1024; off <<= 1) {
            int t = (threadIdx.x >= off) ? buf[threadIdx.x - off] : 0;
            __syncthreads();
            buf[threadIdx.x] += t;
            __syncthreads();
        }
        int out = buf[threadIdx.x] + carry;
        if (i < len) a[i] = out;
        __syncthreads();                      // everyone has read carry
        if (threadIdx.x == 1023) carry = out; // chunk total
        __syncthreads();
    }
}

// dinv[i] = rsqrt(in_degree + 1 self loop); cursor = row start for the scatter
__global__ void k_dinv_cursor(const int* __restrict__ rowptr, float* __restrict__ dinv,
                              int* __restrict__ cursor, int n) {
    int i = blockIdx.x * blockDim.x + threadIdx.x;
    if (i < n) {
        int beg = rowptr[i], end = rowptr[i + 1];
        dinv[i] = rsqrtf((float)(end - beg + 1));
        cursor[i] = beg;
    }
}

__global__ void k_fill(const long long* __restrict__ src, const long long* __restrict__ dst,
                       int* __restrict__ cursor, int* __restrict__ esrc, int e) {
    int i = blockIdx.x * blockDim.x + threadIdx.x;
    if (i < e) {
        int d = (int)dst[i];
        int pos = atomicAdd(&cursor[d], 1);
        esrc[pos] = (int)src[i];
    }
}

// ================= WMMA GEMM: H[n,128] = X[n,128] @ W[128,128] =================
// W (64KB) is staged to LDS via async global->LDS DMA (ASYNCcnt), then each wave
// computes one 16x16 tile with 32x v_wmma_f32_16x16x4_f32 (fp32-exact).
__global__ __launch_bounds__(256) void k_gemm(const float* __restrict__ X,
                                              const float* __restrict__ W,
                                              float* __restrict__ H, int n) {
    __shared__ float Ws[D * D];                         // 64 KB
    {
        // 256 threads x 16 iterations x 16B = 64KB, async copy to LDS
        for (int it = 0; it < 16; ++it) {
            int fidx = (it * 256 + (int)threadIdx.x) * 4;               // float index
            unsigned int  ldsa = (unsigned int)(unsigned long long)(uintptr_t)&Ws[fidx];
            unsigned long long ga = (unsigned long long)(uintptr_t)(W + fidx);
            asm volatile("global_load_async_to_lds_b128 %0, %1, off"
                         :: "v"(ldsa), "v"(ga) : "memory");
        }
        asm volatile("s_wait_asynccnt 0x0" ::: "memory");
        __syncthreads();
    }

    const int wave = threadIdx.x >> 5;
    const int lane = threadIdx.x & 31;
    const int ll   = lane & 15;     // M (A) / N (B) within tile
    const int lh   = lane >> 4;     // K pair select {0,1} vs {2,3}
    const int m0   = blockIdx.x * 16;
    const int n0   = wave * 16;

    int rowA = m0 + ll;
    if (rowA >= n) rowA = n - 1;                        // clamp reads; EXEC stays all-ones
    const float* __restrict__ xrow = X + (size_t)rowA * D;
    const float* __restrict__ wcol = Ws + 2 * lh * D + n0 + ll;

    v8f acc = {};
#pragma unroll 8
    for (int k0 = 0; k0 < D; k0 += 4) {
        v2f a = *(const v2f*)(xrow + k0 + 2 * lh);      // A: lane=M, regs=K pair
        v2f b;
        b.x = wcol[k0 * D];                             // B: lane=N, regs=K pair (LDS)
        b.y = wcol[k0 * D + D];
        acc = __builtin_amdgcn_wmma_f32_16x16x4_f32(false, a, false, b,
                                                    (short)0, acc, false, false);
    }

    // C/D: VGPR v -> rows {m0+v, m0+v+8}, col = n0+ll. Wave-uniform guard only.
    float* __restrict__ hout = H + (size_t)(m0 + 8 * lh) * D + n0 + ll;
    if (m0 + 16 <= n) {
#pragma unroll
        for (int v = 0; v < 8; ++v) hout[(size_t)v * D] = acc[v];
    } else {
#pragma unroll
        for (int v = 0; v < 8; ++v) {
            if (m0 + 8 * lh + v < n) hout[(size_t)v * D] = acc[v];
        }
    }
}

// ====== aggregation (CSR gather, atomic-free): one wave per destination node ======
// out[i] = dinv[i] * ( dinv[i]*h[i] + sum_j dinv[src_j]*h[src_j] ) + bias
__global__ __launch_bounds__(256) void k_gather_agg(const float* __restrict__ H,
                                                    const float* __restrict__ dinv,
                                                    const int* __restrict__ rowptr,
                                                    const int* __restrict__ esrc,
                                                    const float* __restrict__ bias,
                                                    float* __restrict__ out, int n) {
    int wid  = __builtin_amdgcn_readfirstlane(threadIdx.x >> 5);  // force SGPR chain
    int node = blockIdx.x * 8 + wid;
    if (node >= n) return;
    int lane = threadIdx.x & 31;
    int beg  = rowptr[node];
    int end  = rowptr[node + 1];
    float di = dinv[node];

    float4 hv = *(const float4*)(H + (size_t)node * D + lane * 4);
    float4 acc;                                   // self loop contributes di*h (x di later)
    acc.x = di * hv.x; acc.y = di * hv.y; acc.z = di * hv.z; acc.w = di * hv.w;

    for (int j = beg; j < end; ++j) {
        int   s = esrc[j];                        // wave-uniform -> scalar loads
        float w = dinv[s];
        float4 h2 = *(const float4*)(H + (size_t)s * D + lane * 4);
        acc.x += w * h2.x; acc.y += w * h2.y; acc.z += w * h2.z; acc.w += w * h2.w;
    }
    float4 bv = *(const float4*)(bias + lane * 4);
    float4 o;
    o.x = di * acc.x + bv.x; o.y = di * acc.y + bv.y;
    o.z = di * acc.z + bv.z; o.w = di * acc.w + bv.w;
    *(float4*)(out + (size_t)node * D + lane * 4) = o;
}

// ================= batchnorm =================
__global__ void k_zero_stat(float* __restrict__ st) { st[threadIdx.x] = 0.0f; }

__global__ __launch_bounds__(128) void k_bn_stats(const float* __restrict__ A,
                                                  float* __restrict__ st, int n) {
    int c = threadIdx.x;                          // one thread per channel, coalesced rows
    float s = 0.0f, q = 0.0f;
    for (int r = blockIdx.x; r < n; r += gridDim.x) {
        float v = A[(size_t)r * D + c];
        s += v; q += v * v;
    }
    atomic_add_f32(&st[c], s);
    atomic_add_f32(&st[D + c], q);
}

__global__ __launch_bounds__(256) void k_bn_apply(const float* __restrict__ A,
                                                  const float* __restrict__ st,
                                                  const float* __restrict__ g,
                                                  const float* __restrict__ b,
                                                  const float* __restrict__ res,
                                                  float* __restrict__ out,
                                                  int n, int hasRes) {
    int t = blockIdx.x * blockDim.x + threadIdx.x;
    if (t >= n * D) return;
    int c = t & (D - 1);
    float inv_n = 1.0f / (float)n;
    float mu  = st[c] * inv_n;
    float var = st[D + c] * inv_n - mu * mu;      // biased variance (ddof=0)
    float rs  = rsqrtf(var + BN_EPS);
    float v   = (A[t] - mu) * rs * g[c] + b[c];
    v = v > 0.0f ? v : 0.0f;
    if (hasRes) v += res[t];
    out[t] = v;
}

// ================= host-side orchestration =================
extern "C" void kernel_launch(void* const* d_in, const int* in_sizes, int n_in,
                              void* d_out, int out_size, void* d_ws, size_t ws_size,
                              hipStream_t stream) {
    const float*     x     = (const float*)d_in[0];
    const long long* ei    = (const long long*)d_in[1];   // int64 per reference
    const float*     W0    = (const float*)d_in[2];
    const float*     b0    = (const float*)d_in[3];
    const float*     Wmid  = (const float*)d_in[4];
    const float*     bmid  = (const float*)d_in[5];
    const float*     Wlast = (const float*)d_in[6];
    const float*     blast = (const float*)d_in[7];
    const float*     gamma = (const float*)d_in[8];
    const float*     beta  = (const float*)d_in[9];

    const int n = in_sizes[0] / D;
    const int e = in_sizes[1] / 2;
    const long long* src = ei;
    const long long* dst = ei + e;

    char* base = (char*)d_ws;
    float* dinv   = (float*)base;  base += sizeof(float) * (size_t)n;
    float* bufA   = (float*)base;  base += sizeof(float) * (size_t)n * D;
    float* bufB   = (float*)base;  base += sizeof(float) * (size_t)n * D;
    float* stat   = (float*)base;  base += sizeof(float) * 256;
    int*   rowptr = (int*)base;    base += sizeof(int) * (size_t)(n + 1);
    int*   cursor = (int*)base;    base += sizeof(int) * (size_t)n;
    int*   esrc   = (int*)base;
    float* out    = (float*)d_out;                 // N*D; also per-layer agg target

    const int TB = 256;

    // ---- build CSR (by dst) + symmetric-normalization coefficients, once per call
    k_zero_i32   <<<(n + 1 + TB - 1) / TB, TB, 0, stream>>>(rowptr, n + 1);
    k_count      <<<(e + TB - 1) / TB, TB, 0, stream>>>(dst, rowptr, e);
    k_scan       <<<1, 1024, 0, stream>>>(rowptr, n + 1);
    k_dinv_cursor<<<(n + TB - 1) / TB, TB, 0, stream>>>(rowptr, dinv, cursor, n);
    k_fill       <<<(e + TB - 1) / TB, TB, 0, stream>>>(src, dst, cursor, esrc, e);

    auto layer = [&](const float* xin, const float* W, const float* bias,
                     const float* g, const float* bt, const float* res,
                     float* hbuf, float* xnext) {
        k_gemm<<<(n + 15) / 16, 256, 0, stream>>>(xin, W, hbuf, n);
        k_gather_agg<<<(n + 7) / 8, 256, 0, stream>>>(hbuf, dinv, rowptr, esrc, bias, out, n);
        if (g) {
            k_zero_stat<<<1, 256, 0, stream>>>(stat);
            k_bn_stats<<<1024, 128, 0, stream>>>(out, stat, n);
            k_bn_apply<<<(n * D + TB - 1) / TB, TB, 0, stream>>>(
                out, stat, g, bt, res, xnext, n, res != nullptr ? 1 : 0);
        }
    };

    // layer 0: no residual; h into bufB, bn output into bufA
    layer(x, W0, b0, gamma, beta, nullptr, bufB, bufA);
    float* cur = bufA;
    float* oth = bufB;
    // 4 middle layers with residual; bn output overwrites the (dead) h buffer
    for (int i = 0; i < 4; ++i) {
        layer(cur, Wmid + (size_t)i * D * D, bmid + (size_t)i * D,
              gamma + (size_t)(i + 1) * D, beta + (size_t)(i + 1) * D,
              cur, oth, oth);
        float* t = cur; cur = oth; oth = t;
    }
    // output layer: no BN/ReLU; aggregation lands directly in d_out
    layer(cur, Wlast, blast, nullptr, nullptr, nullptr, oth, nullptr);
}